// GraphRNA_46093589020846
// MI455X (gfx1250) — compile-verified
//
#include <hip/hip_runtime.h>

#define HID 128
#define NS  20000
#define NM  100000
#define ES  1000000
#define EM  2000000
#define EL  500000
#define NLAYERS 2

typedef float v2f __attribute__((ext_vector_type(2)));
typedef float v8f __attribute__((ext_vector_type(8)));

// ---------------------------------------------------------------- utilities
__global__ __launch_bounds__(256) void k_zero(float* __restrict__ p, int n) {
  int i = blockIdx.x * 256 + threadIdx.x;
  if (i < n) p[i] = 0.0f;
}

// one thread per edge: histogram degree at src and dst
__global__ __launch_bounds__(256) void k_count(const int* __restrict__ src,
                                               const int* __restrict__ dst,
                                               float* __restrict__ cnt_src,
                                               float* __restrict__ cnt_dst, int E) {
  int e = blockIdx.x * 256 + threadIdx.x;
  if (e < E) {
    unsafeAtomicAdd(cnt_src + src[e], 1.0f);
    unsafeAtomicAdd(cnt_dst + dst[e], 1.0f);
  }
}

// in-place: c -> 1/max(c,1)   (SAGE mean divisor)
__global__ __launch_bounds__(256) void k_inv_cnt(float* __restrict__ p, int n) {
  int i = blockIdx.x * 256 + threadIdx.x;
  if (i < n) p[i] = 1.0f / fmaxf(p[i], 1.0f);
}

// in-place: d -> rsqrt(d + 1)  (GCN sym-norm with self loop)
__global__ __launch_bounds__(256) void k_dinv(float* __restrict__ p, int n) {
  int i = blockIdx.x * 256 + threadIdx.x;
  if (i < n) p[i] = rsqrtf(p[i] + 1.0f);
}

// repack W[128][128] (row-major K x N) into k-pair interleaved layout:
// Wpk[(k>>1)*256 + n*2 + (k&1)] = W[k*128 + n]
// -> B fragment (W[k][col], W[k+1][col]) becomes one aligned 8-byte load.
__global__ __launch_bounds__(256) void k_repack(const float* __restrict__ W,
                                                float* __restrict__ Wpk) {
  int i = blockIdx.x * 256 + threadIdx.x;   // 0 .. 16383
  int k = i >> 7;
  int n = i & 127;
  Wpk[(size_t)(k >> 1) * 256 + n * 2 + (k & 1)] = W[i];
}

// ------------------------------------------------------------- edge scatter
// one wave32 per edge; lane handles 4 of 128 feats (float4 = B128 path).
// dinv == nullptr -> weight 1 (SAGE sum); else weight = dinv[s]*dinv[d] (GCN).
__global__ __launch_bounds__(256) void k_scatter(const int* __restrict__ src,
                                                 const int* __restrict__ dst,
                                                 const float* __restrict__ x,
                                                 const float* __restrict__ dinv,
                                                 float* __restrict__ out, int E) {
  int e = blockIdx.x * 8 + (threadIdx.x >> 5);
  if (e >= E) return;
  int lane = threadIdx.x & 31;
  int s = src[e], d = dst[e];
  float w = dinv ? dinv[s] * dinv[d] : 1.0f;
  const float4 v = *(const float4*)(x + (size_t)s * HID + lane * 4);
  float* o = out + (size_t)d * HID + lane * 4;
  unsafeAtomicAdd(o + 0, w * v.x);
  unsafeAtomicAdd(o + 1, w * v.y);
  unsafeAtomicAdd(o + 2, w * v.z);
  unsafeAtomicAdd(o + 3, w * v.w);
}

// GCN self-loop term: out[i,:] += dinv[i]^2 * xw[i,:]
__global__ __launch_bounds__(256) void k_selfadd(float* __restrict__ out,
                                                 const float* __restrict__ x,
                                                 const float* __restrict__ dinv,
                                                 int total) {
  int i = blockIdx.x * 256 + threadIdx.x;
  if (i < total) {
    float s = dinv[i >> 7];
    out[i] += s * s * x[i];
  }
}

// ------------------------------------------------------------- WMMA GEMM
// out[16 x 128 strip] (+)= (rowscale * A[16 x 128]) @ W[128 x 128] (+bias)(relu)
// Block = 256 threads = 8 waves; wave w computes the 16x16 tile at cols [16w,16w+16).
// Uses V_WMMA_F32_16X16X4_F32 (exact fp32, matching reference numerics).
// Wpk is the k-pair repacked weight (see k_repack).
__global__ __launch_bounds__(256) void k_gemm(const float* __restrict__ A,
                                              const float* __restrict__ rowscale,
                                              const float* __restrict__ Wpk,
                                              float* __restrict__ out,
                                              const float* __restrict__ b0,
                                              const float* __restrict__ b1,
                                              const float* __restrict__ b2,
                                              int accumulate, int relu) {
  __shared__ float As[16][132];  // pad 128->132 floats: banks 4m+k, conflict-free
  const int tid  = threadIdx.x;
  const int lane = tid & 31;
  const int wave = tid >> 5;
  const int row0 = blockIdx.x * 16;

  // cooperative A-strip load (2048 floats = 256 thr * 2 float4), rowscale fused
#pragma unroll
  for (int i = 0; i < 2; ++i) {
    int idx = tid + i * 256;      // 0..511 float4 chunks
    int r   = idx >> 5;           // 16 rows
    int c4  = idx & 31;           // 32 float4 per row
    float4 v = *(const float4*)(A + (size_t)(row0 + r) * HID + c4 * 4);
    float  s = rowscale ? rowscale[row0 + r] : 1.0f;
    As[r][c4 * 4 + 0] = v.x * s;
    As[r][c4 * 4 + 1] = v.y * s;
    As[r][c4 * 4 + 2] = v.z * s;
    As[r][c4 * 4 + 3] = v.w * s;
  }
  __syncthreads();

  // ISA layouts (7.12.2): A 16x4 f32: lanes 0-15 hold M=lane,K={k0,k0+1};
  // lanes 16-31 hold M=lane-16,K={k0+2,k0+3}. B 4x16: VGPR0 = K-rows {0|2},
  // VGPR1 = K-rows {1|3}, N = lane&15.
  const int m   = lane & 15;
  const int kk  = (lane >> 4) << 1;     // 0 or 2
  const int col = wave * 16 + m;
  // per-lane base into packed W: k-pair index = (k0+kk)>>1, element = col*2
  const float* Wbase = Wpk + (size_t)(kk >> 1) * 256 + col * 2;

  v8f c = {};
#pragma unroll 8
  for (int k0 = 0; k0 < HID; k0 += 4) {
    v2f a;
    a.x = As[m][k0 + kk];
    a.y = As[m][k0 + kk + 1];
    v2f b = *(const v2f*)(Wbase + (size_t)(k0 >> 1) * 256);  // one b64 load
    c = __builtin_amdgcn_wmma_f32_16x16x4_f32(false, a, false, b, (short)0, c,
                                              false, false);
  }

  float bias = 0.0f;
  if (b0) bias += b0[col];
  if (b1) bias += b1[col];
  if (b2) bias += b2[col];

  // C/D layout: VGPR r -> row M = r + 8*(lane>=16), col = lane&15
  const int mh = (lane >> 4) << 3;
#pragma unroll
  for (int r = 0; r < 8; ++r) {
    size_t o   = (size_t)(row0 + mh + r) * HID + col;
    float  val = c[r] + bias;
    if (accumulate) val += out[o];
    if (relu) val = fmaxf(val, 0.0f);
    out[o] = val;
  }
}

// ------------------------------------------------------------- classifier
// one wave32 per labeled pair: 128-dim dot product, wave reduce, lane0 writes.
__global__ __launch_bounds__(256) void k_classify(const int* __restrict__ ls,
                                                  const int* __restrict__ ld,
                                                  const float* __restrict__ xs,
                                                  const float* __restrict__ xm,
                                                  float* __restrict__ out, int E) {
  int e = blockIdx.x * 8 + (threadIdx.x >> 5);
  if (e >= E) return;
  int lane = threadIdx.x & 31;
  float4 a = *(const float4*)(xs + (size_t)ls[e] * HID + lane * 4);
  float4 b = *(const float4*)(xm + (size_t)ld[e] * HID + lane * 4);
  float d = a.x * b.x + a.y * b.y + a.z * b.z + a.w * b.w;
#pragma unroll
  for (int off = 16; off > 0; off >>= 1) d += __shfl_down(d, off, 32);
  if (lane == 0) out[e] = d;
}

// ---------------------------------------------------------------- driver
extern "C" void kernel_launch(void* const* d_in, const int* in_sizes, int n_in,
                              void* d_out, int out_size, void* d_ws, size_t ws_size,
                              hipStream_t stream) {
  const int* s2m_src = (const int*)d_in[0];
  const int* s2m_dst = (const int*)d_in[1];
  const int* m2m_src = (const int*)d_in[2];
  const int* m2m_dst = (const int*)d_in[3];
  const int* lbl_src = (const int*)d_in[4];
  const int* lbl_dst = (const int*)d_in[5];
  const float* srna  = (const float*)d_in[6];
  const float* mrna  = (const float*)d_in[7];
  const float* Wl_s2m = (const float*)d_in[8];
  const float* bl_s2m = (const float*)d_in[9];
  const float* Wr_s2m = (const float*)d_in[10];
  const float* Wl_rev = (const float*)d_in[11];
  const float* bl_rev = (const float*)d_in[12];
  const float* Wr_rev = (const float*)d_in[13];
  const float* Wg_f   = (const float*)d_in[14];
  const float* bg_f   = (const float*)d_in[15];
  const float* Wg_r   = (const float*)d_in[16];
  const float* bg_r   = (const float*)d_in[17];

  // workspace layout (floats)
  float* ws = (float*)d_ws;
  float* xs_buf0 = ws;                                   // NS*HID
  float* xs_buf1 = xs_buf0 + (size_t)NS * HID;           // NS*HID
  float* xm_buf0 = xs_buf1 + (size_t)NS * HID;           // NM*HID
  float* xm_buf1 = xm_buf0 + (size_t)NM * HID;           // NM*HID
  float* tmp     = xm_buf1 + (size_t)NM * HID;           // NM*HID (xw / agg reuse)
  float* inv_cnt_m = tmp + (size_t)NM * HID;             // NM
  float* inv_cnt_s = inv_cnt_m + NM;                     // NS
  float* dinv_f    = inv_cnt_s + NS;                     // NM
  float* dinv_r    = dinv_f + NM;                        // NM
  float* wpk       = dinv_r + NM;                        // HID*HID packed weights

  float* xs_bufs[2] = {xs_buf0, xs_buf1};
  float* xm_bufs[2] = {xm_buf0, xm_buf1};

#define ZERO(p, n) k_zero<<<((n) + 255) / 256, 256, 0, stream>>>((p), (int)(n))
  // repack weights then GEMM (stream-ordered, capture-safe)
#define GEMM(Amat, rs, Wmat, Out, B0, B1, B2, acc, rl, nrows)                      \
  do {                                                                             \
    k_repack<<<(HID * HID) / 256, 256, 0, stream>>>((Wmat), wpk);                  \
    k_gemm<<<(nrows) / 16, 256, 0, stream>>>((Amat), (rs), wpk, (Out), (B0), (B1), \
                                             (B2), (acc), (rl));                   \
  } while (0)

  // ---- edge-invariant degree precompute ----
  ZERO(inv_cnt_m, NM);
  ZERO(inv_cnt_s, NS);
  ZERO(dinv_f, NM);
  ZERO(dinv_r, NM);
  k_count<<<(ES + 255) / 256, 256, 0, stream>>>(s2m_src, s2m_dst, inv_cnt_s, inv_cnt_m, ES);
  k_count<<<(EM + 255) / 256, 256, 0, stream>>>(m2m_src, m2m_dst, dinv_r, dinv_f, EM);
  k_inv_cnt<<<(NM + 255) / 256, 256, 0, stream>>>(inv_cnt_m, NM);
  k_inv_cnt<<<(NS + 255) / 256, 256, 0, stream>>>(inv_cnt_s, NS);
  k_dinv<<<(NM + 255) / 256, 256, 0, stream>>>(dinv_f, NM);
  k_dinv<<<(NM + 255) / 256, 256, 0, stream>>>(dinv_r, NM);

  const float* xs_cur = srna;
  const float* xm_cur = mrna;

  for (int l = 0; l < NLAYERS; ++l) {
    float* xs_nxt = xs_bufs[l];
    float* xm_nxt = xm_bufs[l];
    const int wo = l * HID * HID;
    const int bo = l * HID;

    // ================= mRNA update (into xm_nxt) =================
    ZERO(xm_nxt, (size_t)NM * HID);

    // GCN forward: xw = x_m @ Wg_f ; scatter norm*xw[src]->dst ; + dinv^2*xw
    GEMM(xm_cur, nullptr, Wg_f + wo, tmp, nullptr, nullptr, nullptr, 0, 0, NM);
    k_scatter<<<EM / 8, 256, 0, stream>>>(m2m_src, m2m_dst, tmp, dinv_f, xm_nxt, EM);
    k_selfadd<<<(NM * HID) / 256, 256, 0, stream>>>(xm_nxt, tmp, dinv_f, NM * HID);

    // GCN reverse (flipped edges)
    GEMM(xm_cur, nullptr, Wg_r + wo, tmp, nullptr, nullptr, nullptr, 0, 0, NM);
    k_scatter<<<EM / 8, 256, 0, stream>>>(m2m_dst, m2m_src, tmp, dinv_r, xm_nxt, EM);
    k_selfadd<<<(NM * HID) / 256, 256, 0, stream>>>(xm_nxt, tmp, dinv_r, NM * HID);

    // SAGE s->m : agg = sum x_s[src] ; += (agg/cnt)@Wl ; += x_m@Wr + biases ; relu
    ZERO(tmp, (size_t)NM * HID);
    k_scatter<<<ES / 8, 256, 0, stream>>>(s2m_src, s2m_dst, xs_cur, nullptr, tmp, ES);
    GEMM(tmp, inv_cnt_m, Wl_s2m + wo, xm_nxt, nullptr, nullptr, nullptr, 1, 0, NM);
    GEMM(xm_cur, nullptr, Wr_s2m + wo, xm_nxt, bl_s2m + bo, bg_f + bo, bg_r + bo, 1, 1, NM);

    // ================= sRNA update (into xs_nxt), uses OLD x_m ===
    ZERO(tmp, (size_t)NS * HID);
    k_scatter<<<ES / 8, 256, 0, stream>>>(s2m_dst, s2m_src, xm_cur, nullptr, tmp, ES);
    GEMM(tmp, inv_cnt_s, Wl_rev + wo, xs_nxt, nullptr, nullptr, nullptr, 0, 0, NS);
    GEMM(xs_cur, nullptr, Wr_rev + wo, xs_nxt, bl_rev + bo, nullptr, nullptr, 1, 1, NS);

    xs_cur = xs_nxt;
    xm_cur = xm_nxt;
  }

  // ---- classifier: per-pair 128-dim dot ----
  k_classify<<<EL / 8, 256, 0, stream>>>(lbl_src, lbl_dst, xs_cur, xm_cur,
                                         (float*)d_out, EL);
#undef GEMM
#undef ZERO
}